// EEGMamba_1382979469610
// MI455X (gfx1250) — compile-verified
//
#include <hip/hip_runtime.h>
#include <hip/hip_bf16.h>
#include <math.h>

// ---------------------------------------------------------------------------
// EEG-Mamba2 forward for MI455X (gfx1250).
// GEMMs on V_WMMA_F32_16X16X4_F32 (fp32 in/acc): model is L2-resident
// (~105MB < 192MB) and latency-bound, so fp32 WMMA is the right point.
// K is a compile-time template param so the WMMA loop is guard-free:
// 2x global_load_b64 + 1x v_wmma per K-step, no EXEC manipulation.
// SSM scan keeps 64-wide state in VGPRs, one thread per (b,h,p) row.
// ---------------------------------------------------------------------------

#define D_MODEL   200
#define D_INNER   400
#define D_STATE   64
#define NHEADS    8
#define HEADDIM   50
#define CONV_DIM  528           // D_INNER + 2*D_STATE
#define D_IN_PROJ 936           // 2*D_INNER + 2*D_STATE + NHEADS
#define BATCH     16
#define SEQ       570           // 19*30
#define ROWS      (BATCH*SEQ)   // 9120
#define KSPEC     104           // 101 padded to multiple of 4
#define EPS       1e-5f

typedef float v2f __attribute__((ext_vector_type(2)));
typedef float v4f __attribute__((ext_vector_type(4)));
typedef float v8f __attribute__((ext_vector_type(8)));

// ---------------------------------------------------------------------------
// WMMA GEMM:  Out[M,N] = A[M,K] @ W[N,K]^T  (+Out if ADDC) (+bias if BIAS)
// K compile-time, K % 4 == 0.  One 16x16 tile per wave, 4 waves per block.
// Reads are clamped (never OOB), stores guarded in epilogue only.
// ---------------------------------------------------------------------------
template<int K, bool ADDC, bool BIAS>
__global__ __launch_bounds__(128)
void gemm_wmma_f32(const float* __restrict__ A, const float* __restrict__ W,
                   const float* __restrict__ bias, float* __restrict__ Out,
                   int M, int N)
{
    const int wid  = threadIdx.x >> 5;
    const int lane = threadIdx.x & 31;
    const int tn   = blockIdx.x * 4 + wid;
    const int tm   = blockIdx.y;
    if (tn * 16 >= N) return;                 // whole-wave uniform exit

    const int half = lane >> 4;               // 0 or 1
    const int l15  = lane & 15;
    const int kb   = half * 2;                // lane's K sub-offset (0 or 2)
    const int mA   = min(tm * 16 + l15, M - 1);   // clamped: reads always legal
    const int nB   = min(tn * 16 + l15, N - 1);

    const float* __restrict__ Arow = A + (size_t)mA * K + kb;
    const float* __restrict__ Wrow = W + (size_t)nB * K + kb;

    __builtin_prefetch(Arow, 0, 3);           // global_prefetch_b8
    __builtin_prefetch(Wrow, 0, 3);

    // C/D layout: VGPR r -> row tm*16 + r + 8*half, col tn*16 + l15
    v8f c = {0.f,0.f,0.f,0.f,0.f,0.f,0.f,0.f};
    if (ADDC) {
#pragma unroll
        for (int r = 0; r < 8; ++r) {
            int row = tm * 16 + r + half * 8;
            int col = tn * 16 + l15;
            if (row < M && col < N) c[r] = Out[(size_t)row * N + col];
        }
    }

#pragma unroll 4
    for (int k0 = 0; k0 < K; k0 += 4) {
        v2f a = *(const v2f*)(Arow + k0);      // global_load_b64
        v2f b = *(const v2f*)(Wrow + k0);      // global_load_b64
        // D = A(16x4) x B(4x16) + C, fp32 accumulate
        c = __builtin_amdgcn_wmma_f32_16x16x4_f32(false, a, false, b,
                                                  (short)0, c, false, false);
    }

#pragma unroll
    for (int r = 0; r < 8; ++r) {
        int row = tm * 16 + r + half * 8;
        int col = tn * 16 + l15;
        if (row < M && col < N) {
            float v = c[r];
            if (BIAS) v += bias[col];
            Out[(size_t)row * N + col] = v;
        }
    }
}

// ---------------------------------------------------------------------------
// Patch embed: time conv (kernel 49, stride 25, pad 24 along feature dim)
// out t[b,o,h,w], 16*25*570*8 elements
// ---------------------------------------------------------------------------
__global__ void time_conv_kernel(const float* __restrict__ x,
                                 const float* __restrict__ w49,
                                 float* __restrict__ t)
{
    int idx = blockIdx.x * blockDim.x + threadIdx.x;
    if (idx >= BATCH * 25 * SEQ * 8) return;
    int wv  = idx & 7;
    int rem = idx >> 3;
    int hp  = rem % SEQ; rem /= SEQ;
    int o   = rem % 25;
    int b   = rem / 25;
    const float* xr = x + ((size_t)b * SEQ + hp) * D_MODEL;
    const float* wk = w49 + o * 49;
    int base = wv * 25 - 24;
    float acc = 0.f;
#pragma unroll
    for (int k = 0; k < 49; ++k) {
        int d = base + k;
        if (d >= 0 && d < D_MODEL) acc += wk[k] * xr[d];
    }
    t[idx] = acc;
}

// ---------------------------------------------------------------------------
// GroupNorm(5 groups of 5 ch over 570x8) + exact GELU, writes transposed into
// temb[b, pos, o*8+w].  One block per (b,group).
// ---------------------------------------------------------------------------
__global__ __launch_bounds__(256)
void groupnorm_gelu_kernel(const float* __restrict__ t,
                           const float* __restrict__ g,
                           const float* __restrict__ bta,
                           float* __restrict__ temb)
{
    __shared__ float s1[256], s2[256];
    const int grp = blockIdx.x % 5;
    const int b   = blockIdx.x / 5;
    const int tid = threadIdx.x;
    const int NEL = 5 * SEQ * 8;              // 22800
    float sum = 0.f, sq = 0.f;
    for (int i = tid; i < NEL; i += 256) {
        int o   = grp * 5 + i / (SEQ * 8);
        int rem = i % (SEQ * 8);
        int hp  = rem / 8, w = rem % 8;
        float v = t[(((size_t)b * 25 + o) * SEQ + hp) * 8 + w];
        sum += v; sq += v * v;
    }
    s1[tid] = sum; s2[tid] = sq; __syncthreads();
    for (int s = 128; s > 0; s >>= 1) {
        if (tid < s) { s1[tid] += s1[tid + s]; s2[tid] += s2[tid + s]; }
        __syncthreads();
    }
    float mu   = s1[0] / (float)NEL;
    float var  = s2[0] / (float)NEL - mu * mu;
    float rstd = rsqrtf(var + EPS);
    for (int i = tid; i < NEL; i += 256) {
        int o   = grp * 5 + i / (SEQ * 8);
        int rem = i % (SEQ * 8);
        int hp  = rem / 8, w = rem % 8;
        float v  = t[(((size_t)b * 25 + o) * SEQ + hp) * 8 + w];
        float xn = (v - mu) * rstd * g[o] + bta[o];
        float ge = 0.5f * xn * (1.f + erff(xn * 0.70710678118f));
        temb[((size_t)b * SEQ + hp) * D_MODEL + o * 8 + w] = ge;
    }
}

// ---------------------------------------------------------------------------
// rDFT magnitude, norm='forward' (/200).  Writes K-padded spec[9120,104]
// (cols 101..103 = 0) so the spec GEMM needs no K tail.
// ---------------------------------------------------------------------------
__global__ void spec_dft_kernel(const float* __restrict__ x,
                                float* __restrict__ spec)
{
    int idx = blockIdx.x * blockDim.x + threadIdx.x;
    if (idx >= ROWS * KSPEC) return;
    int f = idx % KSPEC;
    int m = idx / KSPEC;
    if (f >= 101) { spec[idx] = 0.f; return; }
    const float* xr = x + (size_t)m * D_MODEL;
    const float w0 = -2.f * 3.14159265358979f * (float)f / (float)D_MODEL;
    float re = 0.f, im = 0.f;
    for (int d = 0; d < D_MODEL; ++d) {
        float sn, cs;
        __sincosf(w0 * (float)d, &sn, &cs);
        re += xr[d] * cs;
        im += xr[d] * sn;
    }
    spec[idx] = sqrtf(re * re + im * im) * (1.f / (float)D_MODEL);
}

// Zero-pad spec_w (200,101) -> (200,104) workspace copy.
__global__ void pad_specw_kernel(const float* __restrict__ w,
                                 float* __restrict__ wp)
{
    int idx = blockIdx.x * blockDim.x + threadIdx.x;
    if (idx >= D_MODEL * KSPEC) return;
    int f = idx % KSPEC;
    int r = idx / KSPEC;
    wp[idx] = (f < 101) ? w[r * 101 + f] : 0.f;
}

// ---------------------------------------------------------------------------
// Depthwise 7x7 positional conv over (19,30), pad 3, + residual add -> hidden
// ---------------------------------------------------------------------------
__global__ void pos_conv_add_kernel(const float* __restrict__ patch,
                                    const float* __restrict__ pw,
                                    float* __restrict__ hidden)
{
    int idx = blockIdx.x * blockDim.x + threadIdx.x;
    if (idx >= ROWS * D_MODEL) return;
    int d   = idx % D_MODEL;
    int rem = idx / D_MODEL;
    int pos = rem % SEQ;
    int b   = rem / SEQ;
    int cc0 = pos / 30, ll0 = pos % 30;
    float acc = patch[idx];
    const float* wk = pw + d * 49;
    for (int i = 0; i < 7; ++i) {
        int cc = cc0 + i - 3;
        if (cc < 0 || cc >= 19) continue;
        for (int j = 0; j < 7; ++j) {
            int ll = ll0 + j - 3;
            if (ll < 0 || ll >= 30) continue;
            acc += wk[i * 7 + j] *
                   patch[((size_t)b * SEQ + cc * 30 + ll) * D_MODEL + d];
        }
    }
    hidden[idx] = acc;
}

// ---------------------------------------------------------------------------
// residual += hidden ; xn = rmsnorm(residual)*w   (D=200, one block per row)
// ---------------------------------------------------------------------------
__global__ __launch_bounds__(64)
void add_rmsnorm_kernel(const float* __restrict__ h, float* __restrict__ res,
                        const float* __restrict__ w, float* __restrict__ xn)
{
    __shared__ float sred[64];
    const int row = blockIdx.x;
    const int tid = threadIdx.x;
    const float* hr = h + (size_t)row * D_MODEL;
    float* rr = res + (size_t)row * D_MODEL;
    float v[4];
    float sq = 0.f;
#pragma unroll
    for (int k = 0; k < 4; ++k) {
        int i = tid + k * 64;
        float t = 0.f;
        if (i < D_MODEL) { t = hr[i] + rr[i]; rr[i] = t; }
        v[k] = t; sq += t * t;
    }
    sred[tid] = sq; __syncthreads();
    for (int s = 32; s > 0; s >>= 1) {
        if (tid < s) sred[tid] += sred[tid + s];
        __syncthreads();
    }
    float rstd = rsqrtf(sred[0] / (float)D_MODEL + EPS);
#pragma unroll
    for (int k = 0; k < 4; ++k) {
        int i = tid + k * 64;
        if (i < D_MODEL) xn[(size_t)row * D_MODEL + i] = v[k] * rstd * w[i];
    }
}

// ---------------------------------------------------------------------------
// dt = softplus(zxbcdt[...,928:936] + dt_bias)
// ---------------------------------------------------------------------------
__global__ void dt_kernel(const float* __restrict__ zx,
                          const float* __restrict__ dtb,
                          float* __restrict__ dt)
{
    int idx = blockIdx.x * blockDim.x + threadIdx.x;
    if (idx >= ROWS * NHEADS) return;
    int h   = idx % NHEADS;
    int row = idx / NHEADS;
    float v = zx[(size_t)row * D_IN_PROJ + (D_INNER + CONV_DIM) + h] + dtb[h];
    dt[idx] = (v > 20.f) ? v : log1pf(__expf(v));
}

// ---------------------------------------------------------------------------
// Causal depthwise conv1d (k=4, left pad 3) over seq, + bias, + SiLU.
// Reads the xBC slice [400:928) straight out of zxbcdt.
// ---------------------------------------------------------------------------
__global__ void conv1d_silu_kernel(const float* __restrict__ zx,
                                   const float* __restrict__ cw,
                                   const float* __restrict__ cb,
                                   float* __restrict__ xBCc)
{
    int idx = blockIdx.x * blockDim.x + threadIdx.x;
    if (idx >= BATCH * SEQ * CONV_DIM) return;
    int ch  = idx % CONV_DIM;
    int rem = idx / CONV_DIM;
    int t   = rem % SEQ;
    int b   = rem / SEQ;
    const float* base = zx + ((size_t)b * SEQ) * D_IN_PROJ + D_INNER + ch;
    const float* wk = cw + ch * 4;
    float acc = cb[ch];
#pragma unroll
    for (int j = 0; j < 4; ++j) {
        int tt = t - 3 + j;
        if (tt >= 0) acc += wk[j] * base[(size_t)tt * D_IN_PROJ];
    }
    acc = acc / (1.f + __expf(-acc));
    xBCc[idx] = acc;
}

// ---------------------------------------------------------------------------
// SSM scan.  One thread per (b,h,p) row; 64-wide state lives in VGPRs.
// Barrier-free; B/C read as float4 broadcasts (global_load_b128 from WGP$).
// ---------------------------------------------------------------------------
__global__ __launch_bounds__(64)
void ssd_scan_kernel(const float* __restrict__ xBC, const float* __restrict__ dt,
                     const float* __restrict__ A_log, const float* __restrict__ Dp,
                     float* __restrict__ y)
{
    const int h = blockIdx.x;      // 0..7
    const int b = blockIdx.y;      // 0..15
    const int p = threadIdx.x;
    if (p >= HEADDIM) return;
    float state[D_STATE];
#pragma unroll
    for (int n = 0; n < D_STATE; ++n) state[n] = 0.f;
    const float Ah = -__expf(A_log[h]);
    const float Dv = Dp[h];
    for (int t = 0; t < SEQ; ++t) {
        const size_t rb  = (size_t)b * SEQ + t;
        const float dtv  = dt[rb * NHEADS + h];
        const float dA   = __expf(dtv * Ah);
        const float xv   = xBC[rb * CONV_DIM + h * HEADDIM + p];
        const float dtx  = dtv * xv;
        const v4f* __restrict__ B4 = (const v4f*)(xBC + rb * CONV_DIM + D_INNER);
        const v4f* __restrict__ C4 = B4 + (D_STATE / 4);
        float acc = 0.f;
#pragma unroll
        for (int q = 0; q < D_STATE / 4; ++q) {
            v4f Bv = B4[q];                    // global_load_b128 (broadcast)
            v4f Cv = C4[q];
#pragma unroll
            for (int j = 0; j < 4; ++j) {
                float s = state[q * 4 + j] * dA + dtx * Bv[j];
                state[q * 4 + j] = s;
                acc += s * Cv[j];
            }
        }
        y[rb * D_INNER + h * HEADDIM + p] = acc + Dv * xv;
    }
}

// ---------------------------------------------------------------------------
// out = rmsnorm(y * silu(z)) * gn_w   (D=400); z is zxbcdt[...,0:400]
// ---------------------------------------------------------------------------
__global__ __launch_bounds__(128)
void gate_rmsnorm_kernel(const float* __restrict__ y, const float* __restrict__ zx,
                         const float* __restrict__ w, float* __restrict__ out)
{
    __shared__ float sred[128];
    const int row = blockIdx.x;
    const int tid = threadIdx.x;
    const float* yr = y  + (size_t)row * D_INNER;
    const float* zr = zx + (size_t)row * D_IN_PROJ;
    float v[4];
    float sq = 0.f;
#pragma unroll
    for (int k = 0; k < 4; ++k) {
        int i = tid + k * 128;
        float t = 0.f;
        if (i < D_INNER) {
            float z = zr[i];
            float s = z / (1.f + __expf(-z));
            t = yr[i] * s;
        }
        v[k] = t; sq += t * t;
    }
    sred[tid] = sq; __syncthreads();
    for (int s = 64; s > 0; s >>= 1) {
        if (tid < s) sred[tid] += sred[tid + s];
        __syncthreads();
    }
    float rstd = rsqrtf(sred[0] / (float)D_INNER + EPS);
#pragma unroll
    for (int k = 0; k < 4; ++k) {
        int i = tid + k * 128;
        if (i < D_INNER) out[(size_t)row * D_INNER + i] = v[k] * rstd * w[i];
    }
}

// ---------------------------------------------------------------------------
extern "C" void kernel_launch(void* const* d_in, const int* in_sizes, int n_in,
                              void* d_out, int out_size, void* d_ws, size_t ws_size,
                              hipStream_t stream)
{
    (void)in_sizes; (void)n_in; (void)out_size; (void)ws_size;

    const float* x          = (const float*)d_in[0];   // (16,19,30,200)
    const float* pe_conv_w  = (const float*)d_in[1];   // (200,1,7,7)
    const float* proj_in_w  = (const float*)d_in[2];   // (25,1,1,49)
    const float* gn_g       = (const float*)d_in[3];   // (25)
    const float* gn_b       = (const float*)d_in[4];   // (25)
    const float* spec_w     = (const float*)d_in[5];   // (200,101)
    const float* norm_w     = (const float*)d_in[6];   // (12,200)
    const float* in_proj_w  = (const float*)d_in[7];   // (12,936,200)
    const float* conv_w     = (const float*)d_in[8];   // (12,528,1,4)
    const float* conv_b     = (const float*)d_in[9];   // (12,528)
    const float* dt_bias    = (const float*)d_in[10];  // (12,8)
    const float* A_log      = (const float*)d_in[11];  // (12,8)
    const float* Dparm      = (const float*)d_in[12];  // (12,8)
    const float* gnorm_w    = (const float*)d_in[13];  // (12,400)
    const float* out_proj_w = (const float*)d_in[14];  // (12,200,400)
    const float* norm_f_w   = (const float*)d_in[15];  // (200)
    const float* head_w     = (const float*)d_in[16];  // (200,200)
    const float* head_b     = (const float*)d_in[17];  // (200)

    float* ws       = (float*)d_ws;
    float* hidden   = ws;                                // 1,824,000 floats
    float* residual = hidden   + (size_t)ROWS * D_MODEL;
    float* xn       = residual + (size_t)ROWS * D_MODEL;
    float* zxbcdt   = xn       + (size_t)ROWS * D_MODEL; // 8,536,320
    float* xBCc     = zxbcdt   + (size_t)ROWS * D_IN_PROJ;
    float* dtb      = xBCc     + (size_t)ROWS * CONV_DIM;
    float* ybuf     = dtb      + (size_t)ROWS * NHEADS;
    float* ygate    = ybuf     + (size_t)ROWS * D_INNER;
    // patch-embed temporaries alias the zxbcdt region (unused until layer 0);
    // all offsets are even float counts -> 8B-aligned for the v2f GEMM loads.
    float* temb     = zxbcdt;                             // 1,824,000
    float* spec     = temb  + (size_t)ROWS * D_MODEL;     //   948,480 (9120x104)
    float* specw    = spec  + (size_t)ROWS * KSPEC;       //    20,800 (200x104)
    float* tconv    = specw + (size_t)D_MODEL * KSPEC;    // 1,824,000

    // ---- patch embed ----
    {
        int n1 = BATCH * 25 * SEQ * 8;
        time_conv_kernel<<<(n1 + 255) / 256, 256, 0, stream>>>(x, proj_in_w, tconv);
        groupnorm_gelu_kernel<<<BATCH * 5, 256, 0, stream>>>(tconv, gn_g, gn_b, temb);
        int n2 = ROWS * KSPEC;
        spec_dft_kernel<<<(n2 + 255) / 256, 256, 0, stream>>>(x, spec);
        int n4 = D_MODEL * KSPEC;
        pad_specw_kernel<<<(n4 + 255) / 256, 256, 0, stream>>>(spec_w, specw);
        // temb += spec @ specw^T   (M=9120, N=200, K=104)
        gemm_wmma_f32<KSPEC, true, false><<<dim3(4, ROWS / 16), 128, 0, stream>>>(
            spec, specw, nullptr, temb, ROWS, D_MODEL);
        int n3 = ROWS * D_MODEL;
        pos_conv_add_kernel<<<(n3 + 255) / 256, 256, 0, stream>>>(temb, pe_conv_w, hidden);
        hipMemsetAsync(residual, 0, (size_t)ROWS * D_MODEL * sizeof(float), stream);
    }

    // ---- 12 Mamba-2 layers ----
    for (int L = 0; L < 12; ++L) {
        add_rmsnorm_kernel<<<ROWS, 64, 0, stream>>>(hidden, residual,
                                                    norm_w + (size_t)L * D_MODEL, xn);
        // zxbcdt = xn @ in_proj^T  (M=9120, N=936, K=200)
        gemm_wmma_f32<D_MODEL, false, false><<<dim3(15, ROWS / 16), 128, 0, stream>>>(
            xn, in_proj_w + (size_t)L * D_IN_PROJ * D_MODEL, nullptr, zxbcdt,
            ROWS, D_IN_PROJ);
        int nd = ROWS * NHEADS;
        dt_kernel<<<(nd + 255) / 256, 256, 0, stream>>>(zxbcdt, dt_bias + L * NHEADS, dtb);
        int nc = BATCH * SEQ * CONV_DIM;
        conv1d_silu_kernel<<<(nc + 255) / 256, 256, 0, stream>>>(
            zxbcdt, conv_w + (size_t)L * CONV_DIM * 4, conv_b + (size_t)L * CONV_DIM, xBCc);
        ssd_scan_kernel<<<dim3(NHEADS, BATCH), 64, 0, stream>>>(
            xBCc, dtb, A_log + L * NHEADS, Dparm + L * NHEADS, ybuf);
        gate_rmsnorm_kernel<<<ROWS, 128, 0, stream>>>(
            ybuf, zxbcdt, gnorm_w + (size_t)L * D_INNER, ygate);
        // hidden = ygate @ out_proj^T  (M=9120, N=200, K=400)
        gemm_wmma_f32<D_INNER, false, false><<<dim3(4, ROWS / 16), 128, 0, stream>>>(
            ygate, out_proj_w + (size_t)L * D_MODEL * D_INNER, nullptr, hidden,
            ROWS, D_MODEL);
    }

    // ---- final norm + head ----
    add_rmsnorm_kernel<<<ROWS, 64, 0, stream>>>(hidden, residual, norm_f_w, xn);
    gemm_wmma_f32<D_MODEL, false, true><<<dim3(4, ROWS / 16), 128, 0, stream>>>(
        xn, head_w, head_b, (float*)d_out, ROWS, D_MODEL);
}